// MultiHeadAttention_15874199126303
// MI455X (gfx1250) — compile-verified
//
#include <hip/hip_runtime.h>
#include <hip/hip_bf16.h>
#include <math.h>

// ---------------------------------------------------------------------------
// MultiHeadAttention forward for MI455X (gfx1250, wave32, WMMA).
// B=2, S=2048, D_MODEL=1024, H=16, DEPTH=64. All math fp32 via
// V_WMMA_F32_16X16X4_F32. Global->LDS staging uses CDNA5 async-to-LDS
// (GLOBAL_LOAD_ASYNC_TO_LDS_B128, ASYNCcnt) with double-buffered tiles so
// HBM fills overlap the WMMA phase.
// ---------------------------------------------------------------------------

typedef __attribute__((ext_vector_type(2))) float v2f;
typedef __attribute__((ext_vector_type(8))) float v8f;

__device__ __forceinline__ v8f wmma_f32_k4(v2f a, v2f b, v8f c) {
  // 8 args: (neg_a, A, neg_b, B, c_mod, C, reuse_a, reuse_b)
  return __builtin_amdgcn_wmma_f32_16x16x4_f32(false, a, false, b, (short)0, c,
                                               false, false);
}

#define LDS_PAD 68  // 64 + 4 floats padding -> conflict-free column reads

// --- CDNA5 async global->LDS (tracked by ASYNCcnt) -------------------------
#define AS3 __attribute__((address_space(3)))

template <typename T>
__device__ __forceinline__ unsigned lds_addr(T* p) {
  // generic -> LDS addrspace cast, then 32-bit LDS byte offset.
  // (widen through size_t so the host pass, where pointers are 64-bit,
  //  doesn't see a narrowing pointer->int cast; device AS(3) ptrs are 32-bit)
  return (unsigned)(size_t)((AS3 T*)p);
}

__device__ __forceinline__ void async_copy_b128(unsigned lds_off,
                                                const float* g) {
  asm volatile("global_load_async_to_lds_b128 %0, %1, off"
               :
               : "v"(lds_off), "v"((unsigned long long)(uintptr_t)g)
               : "memory");
}

__device__ __forceinline__ void wait_async0() {
  asm volatile("s_wait_asynccnt 0x0" ::: "memory");
}

// ---------------------------------------------------------------------------
// Generic fp32 GEMM + bias: C[M,N] = A[M,K] * W[K,N] + bias[N]
// Block: 256 threads (8 waves), 64x64 output tile, double-buffered async LDS.
// Wave w -> column subtile (w&3), row subtiles at 32*(w>>2) and +16.
// ---------------------------------------------------------------------------
__global__ void __launch_bounds__(256)
gemm_bias_kernel(const float* __restrict__ A, const float* __restrict__ W,
                 const float* __restrict__ bias, float* __restrict__ C,
                 int M, int N, int K) {
  __shared__ float As[2][64][LDS_PAD];
  __shared__ float Ws[2][64][LDS_PAD];

  const int tid  = threadIdx.x;
  const int lane = tid & 31;
  const int wave = tid >> 5;
  const int lrow = lane & 15;
  const int lhi  = lane >> 4;  // 0/1: K-pair half (A/B), M half (C/D)

  const int m0 = blockIdx.y * 64;
  const int n0 = blockIdx.x * 64;
  const int tc  = (wave & 3) * 16;
  const int tr0 = (wave >> 2) * 32;

  const int ldr = tid >> 4;        // 0..15
  const int ldc = (tid & 15) * 4;  // 0..60

  v8f acc0 = {};
  v8f acc1 = {};

  // prime buffer 0
#pragma unroll
  for (int rr = 0; rr < 64; rr += 16) {
    const int r = ldr + rr;
    async_copy_b128(lds_addr(&As[0][r][ldc]), &A[(size_t)(m0 + r) * K + ldc]);
    async_copy_b128(lds_addr(&Ws[0][r][ldc]), &W[(size_t)r * N + (n0 + ldc)]);
  }
  wait_async0();
  __syncthreads();

  for (int k0 = 0; k0 < K; k0 += 64) {
    const int cur = (k0 >> 6) & 1;
    const int nxt = cur ^ 1;

    if (k0 + 64 < K) {  // prefetch next chunk into the other buffer
      const int kn = k0 + 64;
#pragma unroll
      for (int rr = 0; rr < 64; rr += 16) {
        const int r = ldr + rr;
        async_copy_b128(lds_addr(&As[nxt][r][ldc]),
                        &A[(size_t)(m0 + r) * K + (kn + ldc)]);
        async_copy_b128(lds_addr(&Ws[nxt][r][ldc]),
                        &W[(size_t)(kn + r) * N + (n0 + ldc)]);
      }
    }

#pragma unroll
    for (int kk = 0; kk < 64; kk += 4) {
      // A frag (16x4): lane half selects K pair {0,1} vs {2,3}
      v2f a0 = *(const v2f*)&As[cur][tr0 + lrow][kk + 2 * lhi];
      v2f a1 = *(const v2f*)&As[cur][tr0 + 16 + lrow][kk + 2 * lhi];
      // B frag (4x16): vgpr j -> K = kk + 2*lhi + j, lane -> column
      v2f b;
      b.x = Ws[cur][kk + 2 * lhi + 0][tc + lrow];
      b.y = Ws[cur][kk + 2 * lhi + 1][tc + lrow];
      acc0 = wmma_f32_k4(a0, b, acc0);
      acc1 = wmma_f32_k4(a1, b, acc1);
    }

    wait_async0();    // next buffer resident in LDS
    __syncthreads();  // all waves done reading 'cur' before it is refilled
  }

  const float bv = bias[n0 + tc + lrow];
#pragma unroll
  for (int r = 0; r < 8; ++r) {
    const int mrow = m0 + tr0 + r + 8 * lhi;  // C layout: M = r + 8*lhi
    const int ncol = n0 + tc + lrow;
    C[(size_t)mrow * N + ncol] = acc0[r] + bv;
    C[(size_t)(mrow + 16) * N + ncol] = acc1[r] + bv;
  }
}

// ---------------------------------------------------------------------------
// Logits: attn[b,h,q,k] = (1/8) * sum_d Qp[b*S+q, h*64+d] * Kp[b*S+k, h*64+d]
// grid = (S/16, B*H), block = 256 (8 waves). Each wave sweeps 256 k-columns.
// ---------------------------------------------------------------------------
__global__ void __launch_bounds__(256)
logits_kernel(const float* __restrict__ Qp, const float* __restrict__ Kp,
              float* __restrict__ attn, int S) {
  __shared__ float qs[16][LDS_PAD];

  const int tid  = threadIdx.x;
  const int lane = tid & 31;
  const int wave = tid >> 5;
  const int lrow = lane & 15;
  const int lhi  = lane >> 4;

  const int q0 = blockIdx.x * 16;
  const int bh = blockIdx.y;
  const int b  = bh >> 4;
  const int h  = bh & 15;

  {  // stage Q tile 16x64 asynchronously (one b128 per thread)
    const int r = tid >> 4;
    const int c = (tid & 15) * 4;
    async_copy_b128(lds_addr(&qs[r][c]),
                    &Qp[(size_t)(b * S + q0 + r) * 1024 + h * 64 + c]);
  }
  wait_async0();
  __syncthreads();

  const float scale = 0.125f;  // 1/sqrt(64)

  for (int t = 0; t < 16; ++t) {
    const int n0 = wave * 256 + t * 16;  // k-column tile base
    const float* kp = &Kp[(size_t)(b * S + n0 + lrow) * 1024 + h * 64];
    v8f acc = {};
#pragma unroll
    for (int kk = 0; kk < 64; kk += 4) {
      v2f a  = *(const v2f*)&qs[lrow][kk + 2 * lhi];
      v2f bf = *(const v2f*)&kp[kk + 2 * lhi];  // K^T frag: contiguous d-pair
      acc = wmma_f32_k4(a, bf, acc);
    }
#pragma unroll
    for (int r = 0; r < 8; ++r) {
      attn[((size_t)bh * S + (q0 + r + 8 * lhi)) * (size_t)S + n0 + lrow] =
          acc[r] * scale;
    }
  }
}

// ---------------------------------------------------------------------------
// Softmax over last axis, in place. One block (256 thr) per row of 2048.
// ---------------------------------------------------------------------------
__global__ void __launch_bounds__(256)
softmax_kernel(float* __restrict__ attn, int S) {
  __shared__ float red[256];
  float* row = attn + (size_t)blockIdx.x * S;
  const int tid = threadIdx.x;

  float x[8];
  float mx = -3.402823466e38f;
#pragma unroll
  for (int j = 0; j < 8; ++j) {
    x[j] = row[tid + j * 256];
    mx = fmaxf(mx, x[j]);
  }
  red[tid] = mx;
  __syncthreads();
  for (int s = 128; s > 0; s >>= 1) {
    if (tid < s) red[tid] = fmaxf(red[tid], red[tid + s]);
    __syncthreads();
  }
  mx = red[0];
  __syncthreads();

  float sum = 0.f;
#pragma unroll
  for (int j = 0; j < 8; ++j) {
    x[j] = expf(x[j] - mx);
    sum += x[j];
  }
  red[tid] = sum;
  __syncthreads();
  for (int s = 128; s > 0; s >>= 1) {
    if (tid < s) red[tid] += red[tid + s];
    __syncthreads();
  }
  const float inv = 1.0f / red[0];
#pragma unroll
  for (int j = 0; j < 8; ++j) row[tid + j * 256] = x[j] * inv;
}

// ---------------------------------------------------------------------------
// Context: concat[b, q, h*64+d] = sum_k attn[b,h,q,k] * Vp[b*S+k, h*64+d]
// grid = (S/16, B*H), block = 128 (4 waves). Wave w owns d-tile w*16.
// Double-buffered async LDS staging of attn (16xK) and V (Kx64) chunks.
// ---------------------------------------------------------------------------
__global__ void __launch_bounds__(128)
context_kernel(const float* __restrict__ attn, const float* __restrict__ Vp,
               float* __restrict__ Cc, int S) {
  __shared__ float as_[2][16][LDS_PAD];
  __shared__ float vs[2][64][LDS_PAD];

  const int tid  = threadIdx.x;
  const int lane = tid & 31;
  const int wave = tid >> 5;  // 0..3
  const int lrow = lane & 15;
  const int lhi  = lane >> 4;

  const int q0 = blockIdx.x * 16;
  const int bh = blockIdx.y;
  const int b  = bh >> 4;
  const int h  = bh & 15;
  const int n0 = wave * 16;  // d subtile

  const int ldr = tid >> 4;        // 0..7
  const int ldc = (tid & 15) * 4;  // 0..60

  v8f acc = {};

  // prime buffer 0 (k0 = 0)
#pragma unroll
  for (int rr = 0; rr < 16; rr += 8) {
    const int r = ldr + rr;
    async_copy_b128(lds_addr(&as_[0][r][ldc]),
                    &attn[((size_t)bh * S + (q0 + r)) * (size_t)S + ldc]);
  }
#pragma unroll
  for (int rr = 0; rr < 64; rr += 8) {
    const int r = ldr + rr;
    async_copy_b128(lds_addr(&vs[0][r][ldc]),
                    &Vp[(size_t)(b * S + r) * 1024 + h * 64 + ldc]);
  }
  wait_async0();
  __syncthreads();

  for (int k0 = 0; k0 < S; k0 += 64) {
    const int cur = (k0 >> 6) & 1;
    const int nxt = cur ^ 1;

    if (k0 + 64 < S) {
      const int kn = k0 + 64;
#pragma unroll
      for (int rr = 0; rr < 16; rr += 8) {
        const int r = ldr + rr;
        async_copy_b128(lds_addr(&as_[nxt][r][ldc]),
                        &attn[((size_t)bh * S + (q0 + r)) * (size_t)S + kn + ldc]);
      }
#pragma unroll
      for (int rr = 0; rr < 64; rr += 8) {
        const int r = ldr + rr;
        async_copy_b128(lds_addr(&vs[nxt][r][ldc]),
                        &Vp[(size_t)(b * S + kn + r) * 1024 + h * 64 + ldc]);
      }
    }

#pragma unroll
    for (int kk = 0; kk < 64; kk += 4) {
      v2f a = *(const v2f*)&as_[cur][lrow][kk + 2 * lhi];
      v2f bf;
      bf.x = vs[cur][kk + 2 * lhi + 0][n0 + lrow];
      bf.y = vs[cur][kk + 2 * lhi + 1][n0 + lrow];
      acc = wmma_f32_k4(a, bf, acc);
    }

    wait_async0();
    __syncthreads();
  }

#pragma unroll
  for (int r = 0; r < 8; ++r) {
    const int q = q0 + r + 8 * lhi;
    Cc[(size_t)(b * S + q) * 1024 + h * 64 + n0 + lrow] = acc[r];
  }
}

// ---------------------------------------------------------------------------
// Host-side launcher
// ---------------------------------------------------------------------------
extern "C" void kernel_launch(void* const* d_in, const int* in_sizes, int n_in,
                              void* d_out, int out_size, void* d_ws,
                              size_t ws_size, hipStream_t stream) {
  (void)in_sizes; (void)n_in; (void)out_size; (void)ws_size;

  constexpr int B = 2, S = 2048, D = 1024;
  constexpr int M = B * S;  // 4096

  const float* q  = (const float*)d_in[0];
  const float* k  = (const float*)d_in[1];
  const float* v  = (const float*)d_in[2];
  const float* wq = (const float*)d_in[3];
  const float* bq = (const float*)d_in[4];
  const float* wk = (const float*)d_in[5];
  const float* bk = (const float*)d_in[6];
  const float* wv = (const float*)d_in[7];
  const float* bv = (const float*)d_in[8];
  const float* wo = (const float*)d_in[9];
  const float* bo = (const float*)d_in[10];

  float* out  = (float*)d_out;
  float* attn = out + (size_t)M * D;  // attention weights region of d_out

  float* ws = (float*)d_ws;
  float* Qp = ws;                  // [M, D]
  float* Kp = Qp + (size_t)M * D;  // [M, D]
  float* Vp = Kp + (size_t)M * D;  // [M, D]
  float* Cc = Vp + (size_t)M * D;  // [M, D] concat(ctx)

  const dim3 gGemm(D / 64, M / 64);  // (16, 64)
  gemm_bias_kernel<<<gGemm, 256, 0, stream>>>(q, wq, bq, Qp, M, D, D);
  gemm_bias_kernel<<<gGemm, 256, 0, stream>>>(k, wk, bk, Kp, M, D, D);
  gemm_bias_kernel<<<gGemm, 256, 0, stream>>>(v, wv, bv, Vp, M, D, D);

  logits_kernel<<<dim3(S / 16, B * 16), 256, 0, stream>>>(Qp, Kp, attn, S);
  softmax_kernel<<<dim3(B * 16 * S), 256, 0, stream>>>(attn, S);
  context_kernel<<<dim3(S / 16, B * 16), 128, 0, stream>>>(attn, Vp, Cc, S);

  gemm_bias_kernel<<<gGemm, 256, 0, stream>>>(Cc, wo, bo, out, M, D, D);
}